// EEGEncoder_47244640256102
// MI455X (gfx1250) — compile-verified
//
#include <hip/hip_runtime.h>
#include <hip/hip_bf16.h>
#include <math.h>

// ---------------------------------------------------------------------------
// CDNA5 (gfx1250) wave32 WMMA, full-fp32 path: V_WMMA_F32_16X16X4_F32
//   A: 16x4 f32  -> 2 VGPR (lanes 0-15: K=kk+{0,1}; lanes16-31: K=kk+{2,3})
//   B: 4x16 f32  -> 2 VGPR (VGPR0: rows K=kk+0 / kk+2 ; VGPR1: rows kk+1 / kk+3)
//   C/D: 16x16 f32 -> 8 VGPR (VGPR j: M=j (lanes0-15) / M=j+8 (lanes16-31), N=lane&15)
// ---------------------------------------------------------------------------
typedef float v2f __attribute__((ext_vector_type(2)));
typedef float v8f __attribute__((ext_vector_type(8)));
typedef unsigned u32x4 __attribute__((ext_vector_type(4)));
typedef int i32x8 __attribute__((ext_vector_type(8)));
typedef int i32x4 __attribute__((ext_vector_type(4)));

__device__ __forceinline__ v8f wmma_f32(v2f a, v2f b, v8f c) {
  return __builtin_amdgcn_wmma_f32_16x16x4_f32(false, a, false, b, (short)0, c,
                                               false, false);
}

__device__ __forceinline__ float sigmoidf_(float x) { return 1.f / (1.f + expf(-x)); }
__device__ __forceinline__ float gelu_exact(float x) {
  return 0.5f * x * (1.f + erff(x * 0.70710678118654752440f));
}
__device__ __forceinline__ float softplusf_(float x) {
  return fmaxf(x, 0.f) + log1pf(expf(-fabsf(x)));  // stable softplus
}
__device__ __forceinline__ int clampi_(int v, int lo, int hi) {
  return v < lo ? lo : (v > hi ? hi : v);
}

// ---------------------------------------------------------------------------
// Implicit-GEMM conv1d, MT M-tiles per wave (64 x 16 output per wave at MT=4).
// out[b, m, t] = bias[m] + sum_{ci,kp} in[b, ci, t*s+kp-pad] * W[m, ci*ksz+kp]
// (ci,kp) tracked incrementally (no int div); padding handled branch-free:
// clamp the address, load unconditionally, cndmask the value to zero.
// ---------------------------------------------------------------------------
template <int MT>
__global__ __launch_bounds__(32)
void conv1d_wmma(const float* __restrict__ in, const float* __restrict__ W,
                 const float* __restrict__ bias, float* __restrict__ out,
                 int Cin, int Tin, int Cout, int Tout, int ksz, int stride, int pad) {
  const int lane = threadIdx.x;
  const int hf   = lane >> 4;
  const int ml   = lane & 15;
  const int t0   = blockIdx.x * 16;
  const int m0   = blockIdx.y * (16 * MT);
  const int bb   = blockIdx.z;
  const int K    = Cin * ksz;
  const float* inb = in + (size_t)bb * Cin * Tin;
  const int t   = t0 + ml;                 // B-matrix column this lane supplies
  const bool tv = (t < Tout);
  const int tb  = t * stride - pad;        // input base position for this column
  // incremental im2col row decode for rows kr0 = kk+2*hf, kr1 = kr0+1
  int kp0 = 2 * hf, ci0 = 0;
  int kp1 = 2 * hf + 1, ci1 = 0;
  // per-tile A row pointers, bumped by 4 each K-chunk
  const float* wr[MT];
#pragma unroll
  for (int mt = 0; mt < MT; ++mt)
    wr[mt] = W + (size_t)(m0 + mt * 16 + ml) * K + 2 * hf;

  v8f acc[MT];
#pragma unroll
  for (int mt = 0; mt < MT; ++mt) acc[mt] = v8f{};

  for (int kk = 0; kk < K; kk += 4) {
    int tin0 = tb + kp0;
    int tin1 = tb + kp1;
    float v0 = inb[(size_t)ci0 * Tin + clampi_(tin0, 0, Tin - 1)];
    float v1 = inb[(size_t)ci1 * Tin + clampi_(tin1, 0, Tin - 1)];
    v2f bv;
    bv.x = (tv && tin0 >= 0 && tin0 < Tin) ? v0 : 0.f;
    bv.y = (tv && tin1 >= 0 && tin1 < Tin) ? v1 : 0.f;
#pragma unroll
    for (int mt = 0; mt < MT; ++mt) {
      v2f a; a.x = wr[mt][0]; a.y = wr[mt][1];
      acc[mt] = wmma_f32(a, bv, acc[mt]);
      wr[mt] += 4;
    }
    kp0 += 4; if (kp0 >= ksz) { kp0 -= ksz; ++ci0; }
    kp1 += 4; if (kp1 >= ksz) { kp1 -= ksz; ++ci1; }
  }
  if (tv) {
    float* ob = out + (size_t)bb * Cout * Tout;
#pragma unroll
    for (int mt = 0; mt < MT; ++mt)
#pragma unroll
      for (int j = 0; j < 8; ++j) {
        int m = m0 + mt * 16 + j + 8 * hf;
        ob[(size_t)m * Tout + t] = acc[mt][j] + bias[m];
      }
  }
}

// ---------------------------------------------------------------------------
// Batched GEMM, MT M-tiles per wave:
//   Out[b, m, n] = act( opt_bias[m] + sum_k W[m,k] * In[b, k, n] )
// W row-major [M,K]; In/Out feature-major, row stride = N. act: 0=id, 1=softplus
// Ragged-N handling is branch-free: clamp the column address; garbage values in
// lanes n>=N stay confined to those lanes' accumulators and are never stored.
// ---------------------------------------------------------------------------
template <int MT>
__global__ __launch_bounds__(32)
void gemm_wmma(const float* __restrict__ W, const float* __restrict__ In,
               const float* __restrict__ bias, float* __restrict__ Out,
               int M, int N, int K, long inBStride, long outBStride, int act) {
  const int lane = threadIdx.x;
  const int hf   = lane >> 4;
  const int ml   = lane & 15;
  const int n0   = blockIdx.x * 16;
  const int m0   = blockIdx.y * (16 * MT);
  const int n = n0 + ml;
  const bool nv = (n < N);
  const int nc = nv ? n : (N - 1);  // clamped column for loads
  const float* bp = In + (size_t)blockIdx.z * inBStride + (size_t)(2 * hf) * N + nc;
  const float* wr[MT];
#pragma unroll
  for (int mt = 0; mt < MT; ++mt)
    wr[mt] = W + (size_t)(m0 + mt * 16 + ml) * K + 2 * hf;

  v8f acc[MT];
#pragma unroll
  for (int mt = 0; mt < MT; ++mt) acc[mt] = v8f{};

  for (int kk = 0; kk < K; kk += 4) {
    v2f bv;
    bv.x = bp[0];
    bv.y = bp[N];
    if (kk + 4 < K) __builtin_prefetch(bp + 4 * (size_t)N, 0, 1);  // global_prefetch_b8
#pragma unroll
    for (int mt = 0; mt < MT; ++mt) {
      v2f a; a.x = wr[mt][0]; a.y = wr[mt][1];
      acc[mt] = wmma_f32(a, bv, acc[mt]);
      wr[mt] += 4;
    }
    bp += 4 * (size_t)N;
  }
  if (nv) {
    float* ob = Out + (size_t)blockIdx.z * outBStride;
#pragma unroll
    for (int mt = 0; mt < MT; ++mt)
#pragma unroll
      for (int j = 0; j < 8; ++j) {
        int m = m0 + mt * 16 + j + 8 * hf;
        float v = acc[mt][j] + (bias ? bias[m] : 0.f);
        if (act == 1) v = softplusf_(v);
        ob[(size_t)m * N + n] = v;
      }
  }
}

// ---------------------------------------------------------------------------
// GroupNorm(1) stats over (C,T) per sample:  mean[b], var[b]
// ---------------------------------------------------------------------------
__global__ __launch_bounds__(256)
void gn_stats(const float* __restrict__ x, float* __restrict__ mean,
              float* __restrict__ var, int CT) {
  const int b = blockIdx.x;
  const int tid = threadIdx.x;
  const float* xb = x + (size_t)b * CT;
  float s = 0.f, ss = 0.f;
  for (int i = tid; i < CT; i += 256) { float v = xb[i]; s += v; ss += v * v; }
  __shared__ float sh[256], sh2[256];
  sh[tid] = s; sh2[tid] = ss;
  __syncthreads();
  for (int off = 128; off > 0; off >>= 1) {
    if (tid < off) { sh[tid] += sh[tid + off]; sh2[tid] += sh2[tid + off]; }
    __syncthreads();
  }
  if (tid == 0) {
    float m = sh[0] / (float)CT;
    mean[b] = m;
    var[b] = sh2[0] / (float)CT - m * m;
  }
}

// Normalize (in-place) + affine + exact GELU
__global__ __launch_bounds__(256)
void gn_gelu(float* __restrict__ x, const float* __restrict__ g,
             const float* __restrict__ be, const float* __restrict__ mean,
             const float* __restrict__ var, int C, int T, long total) {
  long i = (long)blockIdx.x * blockDim.x + threadIdx.x;
  long stride = (long)gridDim.x * blockDim.x;
  for (; i < total; i += stride) {
    long ct = (long)C * T;
    int b = (int)(i / ct);
    int cch = (int)((i / T) % C);
    float rs = rsqrtf(var[b] + 1e-5f);
    float v = (x[i] - mean[b]) * rs * g[cch] + be[cch];
    x[i] = gelu_exact(v);
  }
}

// ---------------------------------------------------------------------------
// Depthwise causal conv (k=4, left pad 3) + SiLU on xm = rows [0,512) of xz
// ---------------------------------------------------------------------------
__global__ __launch_bounds__(256)
void dwconv_silu(const float* __restrict__ xz, const float* __restrict__ cw,
                 const float* __restrict__ cb, float* __restrict__ xt, long total) {
  long i = (long)blockIdx.x * blockDim.x + threadIdx.x;
  long stride = (long)gridDim.x * blockDim.x;
  for (; i < total; i += stride) {
    int l = (int)(i % 125);
    long bd = i / 125;
    int d = (int)(bd % 512);
    int b = (int)(bd / 512);
    const float* xm = xz + ((size_t)b * 1024 + d) * 125;
    float acc = cb[d];
#pragma unroll
    for (int j = 0; j < 4; ++j) {
      int li = l - 3 + j;
      if (li >= 0) acc += xm[li] * cw[d * 4 + j];
    }
    xt[i] = acc * sigmoidf_(acc);
  }
}

// ---------------------------------------------------------------------------
// Selective scan. One thread per (b,d); 16-state recurrence in registers.
// The 32x125 fp32 slab of proj rows [16,48) (B_t, C_t for this sample) is
// DMA'd into LDS once by the Tensor Data Mover (wave 0 issues, waits on
// TENSORcnt), then every per-step read is a same-address LDS broadcast and
// the main loop runs barrier-free.
// ---------------------------------------------------------------------------
__global__ __launch_bounds__(256)
void mamba_scan(const float* __restrict__ dtf, const float* __restrict__ xt,
                const float* __restrict__ proj, const float* __restrict__ xz,
                const float* __restrict__ A_log, const float* __restrict__ Dp,
                float* __restrict__ y) {
  const int b = blockIdx.x >> 1;
  const int d = ((blockIdx.x & 1) << 8) + threadIdx.x;
  __shared__ float sProj[32 * 125];  // rows 16..47 of proj[b] : B (16) then C (16)

  if (threadIdx.x < 32) {  // wave 0 only: TDM issues per-wave (EXEC ignored)
    unsigned long long ga =
        (unsigned long long)(proj + (size_t)b * 48 * 125 + 16 * 125);
    unsigned lds = (unsigned)(size_t)(void*)&sProj[0];
    u32x4 g0;
    g0[0] = 1u;                                   // count=1, user descriptor
    g0[1] = lds;                                  // lds_addr
    g0[2] = (unsigned)(ga & 0xFFFFFFFFu);         // global_addr[31:0]
    g0[3] = (unsigned)((ga >> 32) & 0x01FFFFFFu)  // global_addr[56:32]
            | (2u << 30);                         // type = 2 ("image")
    i32x8 g1;
    g1[0] = (2 << 16);      // data_size = 4B
    g1[1] = (125 << 16);    // tensor_dim0 = 125 (bits 79:48 low half)
    g1[2] = (32 << 16);     // tensor_dim0 hi = 0 ; tensor_dim1 = 32
    g1[3] = (125 << 16);    // tensor_dim1 hi = 0 ; tile_dim0 = 125
    g1[4] = 32;             // tile_dim1 = 32, tile_dim2 = 0
    g1[5] = 125;            // tensor_dim0_stride = 125
    g1[6] = 0;
    g1[7] = 0;
    i32x4 gz = {0, 0, 0, 0};
#if __clang_major__ >= 23
    i32x8 gz8 = {0, 0, 0, 0, 0, 0, 0, 0};
    __builtin_amdgcn_tensor_load_to_lds(g0, g1, gz, gz, gz8, 0);
#else
    __builtin_amdgcn_tensor_load_to_lds(g0, g1, gz, gz, 0);
#endif
    __builtin_amdgcn_s_wait_tensorcnt(0);
  }
  __syncthreads();

  const float* dtp = dtf + ((size_t)b * 512 + d) * 125;
  const float* xp  = xt  + ((size_t)b * 512 + d) * 125;
  const float* zp  = xz  + ((size_t)b * 1024 + 512 + d) * 125;
  float* yp        = y   + ((size_t)b * 512 + d) * 125;
  float A[16], h[16];
#pragma unroll
  for (int s = 0; s < 16; ++s) { A[s] = -expf(A_log[d * 16 + s]); h[s] = 0.f; }
  const float Dd = Dp[d];

  for (int l = 0; l < 125; ++l) {
    float dt = dtp[l], xv = xp[l];
    float dx = dt * xv, acc = 0.f;
#pragma unroll
    for (int s = 0; s < 16; ++s) {
      h[s] = expf(dt * A[s]) * h[s] + dx * sProj[s * 125 + l];
      acc += h[s] * sProj[(16 + s) * 125 + l];
    }
    float zv = zp[l];
    yp[l] = (acc + xv * Dd) * (zv * sigmoidf_(zv));
  }
}

// ---------------------------------------------------------------------------
// Exact AdaptiveAvgPool1d(125 -> 75), feature-major
// ---------------------------------------------------------------------------
__global__ __launch_bounds__(256)
void pool_kernel(const float* __restrict__ mo, float* __restrict__ pooled, long total) {
  long i = (long)blockIdx.x * blockDim.x + threadIdx.x;
  long stride = (long)gridDim.x * blockDim.x;
  for (; i < total; i += stride) {
    int t = (int)(i % 75);
    long bc = i / 75;
    int s = (t * 125) / 75;
    int e = ((t + 1) * 125 + 74) / 75;  // ceil
    const float* row = mo + bc * 125;
    float acc = 0.f;
    for (int j = s; j < e; ++j) acc += row[j];
    pooled[i] = acc / (float)(e - s);
  }
}

// ---------------------------------------------------------------------------
static inline int cdiv(int a, int b) { return (a + b - 1) / b; }

extern "C" void kernel_launch(void* const* d_in, const int* in_sizes, int n_in,
                              void* d_out, int out_size, void* d_ws, size_t ws_size,
                              hipStream_t stream) {
  (void)in_sizes; (void)n_in; (void)out_size; (void)ws_size;
  const float* x    = (const float*)d_in[0];
  const float* w1   = (const float*)d_in[1];
  const float* b1   = (const float*)d_in[2];
  const float* g1   = (const float*)d_in[3];
  const float* be1  = (const float*)d_in[4];
  const float* w2   = (const float*)d_in[5];
  const float* b2   = (const float*)d_in[6];
  const float* g2   = (const float*)d_in[7];
  const float* be2  = (const float*)d_in[8];
  const float* w3   = (const float*)d_in[9];
  const float* b3   = (const float*)d_in[10];
  const float* g3   = (const float*)d_in[11];
  const float* be3  = (const float*)d_in[12];
  const float* w4   = (const float*)d_in[13];
  const float* b4   = (const float*)d_in[14];
  const float* g4   = (const float*)d_in[15];
  const float* be4  = (const float*)d_in[16];
  const float* in_proj_w = (const float*)d_in[17];
  const float* conv_w    = (const float*)d_in[18];
  const float* conv_b    = (const float*)d_in[19];
  const float* x_proj_w  = (const float*)d_in[20];
  const float* dt_proj_w = (const float*)d_in[21];
  const float* dt_proj_b = (const float*)d_in[22];
  const float* A_log     = (const float*)d_in[23];
  const float* Dp        = (const float*)d_in[24];
  const float* m_out_w   = (const float*)d_in[25];
  const float* wo        = (const float*)d_in[26];
  const float* bo        = (const float*)d_in[27];
  float* out = (float*)d_out;

  const int Bsz = 128;
  float* ws = (float*)d_ws;
  // region A (16.384M f): h1 -> h3 -> xz ; region B: h2 -> h4 -> {xt, dtf}
  float* regA   = ws;
  float* regB   = ws + 16384000;
  float* dtf    = regB + 8192000;
  float* proj   = ws + 32768000;   //  768000 f
  float* ybuf   = ws + 33536000;   // 8192000 f
  float* mo     = ws + 41728000;   // 4096000 f
  float* pooled = ws + 45824000;   // 2457600 f
  float* meanb  = ws + 48281600;   // 128 f
  float* varb   = meanb + 128;     // 128 f

  // ---- conv stack: conv -> GN stats -> GN+GELU (in place) ----
  conv1d_wmma<4><<<dim3(cdiv(2000, 16), 64 / 64, Bsz), 32, 0, stream>>>(
      x, w1, b1, regA, 32, 4000, 64, 2000, 7, 2, 3);
  gn_stats<<<Bsz, 256, 0, stream>>>(regA, meanb, varb, 64 * 2000);
  gn_gelu<<<2048, 256, 0, stream>>>(regA, g1, be1, meanb, varb, 64, 2000,
                                    (long)Bsz * 64 * 2000);
  conv1d_wmma<4><<<dim3(cdiv(1000, 16), 128 / 64, Bsz), 32, 0, stream>>>(
      regA, w2, b2, regB, 64, 2000, 128, 1000, 5, 2, 2);
  gn_stats<<<Bsz, 256, 0, stream>>>(regB, meanb, varb, 128 * 1000);
  gn_gelu<<<2048, 256, 0, stream>>>(regB, g2, be2, meanb, varb, 128, 1000,
                                    (long)Bsz * 128 * 1000);
  conv1d_wmma<4><<<dim3(cdiv(500, 16), 256 / 64, Bsz), 32, 0, stream>>>(
      regB, w3, b3, regA, 128, 1000, 256, 500, 5, 2, 2);
  gn_stats<<<Bsz, 256, 0, stream>>>(regA, meanb, varb, 256 * 500);
  gn_gelu<<<2048, 256, 0, stream>>>(regA, g3, be3, meanb, varb, 256, 500,
                                    (long)Bsz * 256 * 500);
  conv1d_wmma<4><<<dim3(cdiv(125, 16), 256 / 64, Bsz), 32, 0, stream>>>(
      regA, w4, b4, regB, 256, 500, 256, 125, 5, 4, 2);
  gn_stats<<<Bsz, 256, 0, stream>>>(regB, meanb, varb, 256 * 125);
  gn_gelu<<<2048, 256, 0, stream>>>(regB, g4, be4, meanb, varb, 256, 125,
                                    (long)Bsz * 256 * 125);

  // ---- Mamba ----
  gemm_wmma<4><<<dim3(cdiv(125, 16), 1024 / 64, Bsz), 32, 0, stream>>>(
      in_proj_w, regB, nullptr, regA, 1024, 125, 256, 256L * 125, 1024L * 125, 0);
  dwconv_silu<<<2048, 256, 0, stream>>>(regA, conv_w, conv_b, regB,
                                        (long)Bsz * 512 * 125);
  gemm_wmma<3><<<dim3(cdiv(125, 16), 1, Bsz), 32, 0, stream>>>(
      x_proj_w, regB, nullptr, proj, 48, 125, 512, 512L * 125, 48L * 125, 0);
  gemm_wmma<4><<<dim3(cdiv(125, 16), 512 / 64, Bsz), 32, 0, stream>>>(
      dt_proj_w, proj, dt_proj_b, dtf, 512, 125, 16, 48L * 125, 512L * 125, 1);
  mamba_scan<<<Bsz * 2, 256, 0, stream>>>(dtf, regB, proj, regA, A_log, Dp, ybuf);
  gemm_wmma<4><<<dim3(cdiv(125, 16), 256 / 64, Bsz), 32, 0, stream>>>(
      m_out_w, ybuf, nullptr, mo, 256, 125, 512, 512L * 125, 256L * 125, 0);

  // ---- pool 125 -> 75, then final projection 256 -> 128 (+bo) ----
  pool_kernel<<<2048, 256, 0, stream>>>(mo, pooled, (long)Bsz * 256 * 75);
  gemm_wmma<4><<<dim3(cdiv(75, 16), 128 / 64, Bsz), 32, 0, stream>>>(
      wo, pooled, bo, out, 128, 75, 256, 256L * 75, 128L * 75, 0);
}